// UpSampleInterpolation_19215683682641
// MI455X (gfx1250) — compile-verified
//
#include <hip/hip_runtime.h>
#include <hip/hip_bf16.h>

typedef __attribute__((ext_vector_type(2))) float v2f;
typedef __attribute__((ext_vector_type(8))) float v8f;

#define Bc    8
#define CD    256
#define CS    512
#define Mc    4096
#define Sc    1024
#define NPTS  4096
#define COUT  256
#define LDW   768          // 3*CD, row stride of conv_w

// ---------------------------------------------------------------------------
// K0: gather xyz coords for dense & sparse points; copy dense_idx to out tail
// dxyz: [B][3][M], sxyz: [B][3][S]
// ---------------------------------------------------------------------------
__global__ void gather_xyz_kernel(const float* __restrict__ pcd,
                                  const int* __restrict__ dense_idx,
                                  const int* __restrict__ sparse_idx,
                                  float* __restrict__ dxyz,
                                  float* __restrict__ sxyz,
                                  int* __restrict__ idx_out) {
    int t = blockIdx.x * blockDim.x + threadIdx.x;
    const int totalD = Bc * Mc;
    const int totalS = Bc * Sc;
    if (t < totalD) {
        int b = t / Mc, m = t % Mc;
        int p = dense_idx[t];
        const float* pb = pcd + (size_t)b * 3 * NPTS;
        float* db = dxyz + (size_t)b * 3 * Mc;
        db[0 * Mc + m] = pb[0 * NPTS + p];
        db[1 * Mc + m] = pb[1 * NPTS + p];
        db[2 * Mc + m] = pb[2 * NPTS + p];
        idx_out[t] = p;                         // second tuple output (bit copy)
    } else if (t < totalD + totalS) {
        int u = t - totalD;
        int b = u / Sc, s = u % Sc;
        int p = sparse_idx[u];
        const float* pb = pcd + (size_t)b * 3 * NPTS;
        float* sb = sxyz + (size_t)b * 3 * Sc;
        sb[0 * Sc + s] = pb[0 * NPTS + p];
        sb[1 * Sc + s] = pb[1 * NPTS + p];
        sb[2 * Sc + s] = pb[2 * NPTS + p];
    }
}

// ---------------------------------------------------------------------------
// K1: 3-NN per dense point over S sparse points; sparse xyz staged in LDS.
// knn_idx / knn_w: [B][M][3]
// ---------------------------------------------------------------------------
__global__ void knn_kernel(const float* __restrict__ dxyz,
                           const float* __restrict__ sxyz,
                           int* __restrict__ knn_idx,
                           float* __restrict__ knn_w) {
    __shared__ float sx[Sc], sy[Sc], sz[Sc];      // 12 KB of the 320 KB WGP LDS
    const int b = blockIdx.y;
    const int tid = threadIdx.x;
    const float* sb = sxyz + (size_t)b * 3 * Sc;
    for (int i = tid; i < Sc; i += blockDim.x) {
        sx[i] = sb[i];
        sy[i] = sb[Sc + i];
        sz[i] = sb[2 * Sc + i];
    }
    __syncthreads();

    const int m = blockIdx.x * blockDim.x + tid;
    const float* db = dxyz + (size_t)b * 3 * Mc;
    const float px = db[m], py = db[Mc + m], pz = db[2 * Mc + m];

    float d0 = 3.4e38f, d1 = 3.4e38f, d2 = 3.4e38f;
    int i0 = 0, i1 = 0, i2 = 0;
    for (int s = 0; s < Sc; ++s) {
        float ddx = px - sx[s], ddy = py - sy[s], ddz = pz - sz[s];
        float d = ddx * ddx + ddy * ddy + ddz * ddz;
        if (d < d0)      { d2 = d1; i2 = i1; d1 = d0; i1 = i0; d0 = d; i0 = s; }
        else if (d < d1) { d2 = d1; i2 = i1; d1 = d;  i1 = s; }
        else if (d < d2) { d2 = d;  i2 = s; }
    }
    float w0 = 1.f / (d0 + 1e-8f);
    float w1 = 1.f / (d1 + 1e-8f);
    float w2 = 1.f / (d2 + 1e-8f);
    float inv = 1.f / (w0 + w1 + w2);
    size_t base = ((size_t)b * Mc + m) * 3;
    knn_idx[base] = i0; knn_idx[base + 1] = i1; knn_idx[base + 2] = i2;
    knn_w[base] = w0 * inv; knn_w[base + 1] = w1 * inv; knn_w[base + 2] = w2 * inv;
}

// ---------------------------------------------------------------------------
// K2/K3: fp32 WMMA GEMM  Y[b] = Wmat(COUT x KDIM, row stride ldw) @ X[b](KDIM x NDIM)
// Each wave computes a 16x32 output tile (2 accumulators) via
// V_WMMA_F32_16X16X4_F32, K stepping by 4; A tile is reused across both
// N-tiles. KDIM/NDIM are compile-time so every k-step load folds its
// (k*NDIM*4)-byte displacement into the global_load signed-24-bit IOFFSET
// (max 255*4096*4 = 4.2 MB < 8 MB) -> no per-iteration address VALU.
// A layout (ISA 7.12.2, 16x4 f32): lane = m + 16*(k/2), vgpr = k%2 -> float2.
// B layout (4x16 f32, dual):       lane = n + 16*(k/2), vgpr = k%2.
// C/D: vgpr v holds row M = v + 8*(lane/16), col N = lane%16.
// 4 waves/block stack in the output-channel dim sharing the same X tiles.
// ---------------------------------------------------------------------------
template <int KDIM, int NDIM>
__global__ void gemm_wmma_f32(const float* __restrict__ Wmat,
                              const float* __restrict__ X,
                              float* __restrict__ Y) {
    const int lane = threadIdx.x & 31;
    const int wave = threadIdx.x >> 5;
    const int b  = blockIdx.z;
    const int n0 = blockIdx.x * 32;
    const int o0 = (blockIdx.y * 4 + wave) * 16;
    const int r  = lane & 15;           // m (for A) / n (for B) within tile
    const int kh = (lane >> 4) << 1;    // k sub-offset: 0 or 2

    // lane-dependent parts folded into base pointers once; k offsets immediate
    const float* Wp  = Wmat + (size_t)(o0 + r) * LDW + kh;
    const float* Xp0 = X + (size_t)b * KDIM * NDIM + (size_t)kh * NDIM + n0 + r;
    const float* Xp1 = Xp0 + 16;

    v8f acc0 = {};
    v8f acc1 = {};
#pragma unroll 8
    for (int k = 0; k < KDIM; k += 4) {
        v2f a = *(const v2f*)(Wp + k);              // W[o0+r][k+kh .. k+kh+1]
        v2f b0, b1;
        b0.x = Xp0[(size_t)k * NDIM];               // X[k+kh  ][n0+r]
        b0.y = Xp0[(size_t)(k + 1) * NDIM];         // X[k+kh+1][n0+r]
        b1.x = Xp1[(size_t)k * NDIM];               // X[k+kh  ][n0+16+r]
        b1.y = Xp1[(size_t)(k + 1) * NDIM];
        acc0 = __builtin_amdgcn_wmma_f32_16x16x4_f32(
            false, a, false, b0, (short)0, acc0, false, false);
        acc1 = __builtin_amdgcn_wmma_f32_16x16x4_f32(
            false, a, false, b1, (short)0, acc1, false, false);
    }

    float* Yp = Y + ((size_t)b * COUT + o0 + (lane >> 4) * 8) * NDIM + n0 + r;
#pragma unroll
    for (int v = 0; v < 8; ++v) {
        Yp[(size_t)v * NDIM] = acc0[v];
        Yp[(size_t)v * NDIM + 16] = acc1[v];
    }
}

// ---------------------------------------------------------------------------
// K4: interpolate-after-conv:  Y[b][o][m] += sum_k w_k * z[b][o][idx_k]
// z is 8 MB -> fully L2 resident gathers.
// ---------------------------------------------------------------------------
__global__ void interp_add_kernel(const float* __restrict__ z,
                                  const int* __restrict__ knn_idx,
                                  const float* __restrict__ knn_w,
                                  float* __restrict__ Y) {
    const int m = blockIdx.x * blockDim.x + threadIdx.x;
    const int o = blockIdx.y;
    const int b = blockIdx.z;
    size_t kb = ((size_t)b * Mc + m) * 3;
    int i0 = knn_idx[kb], i1 = knn_idx[kb + 1], i2 = knn_idx[kb + 2];
    float w0 = knn_w[kb], w1 = knn_w[kb + 1], w2 = knn_w[kb + 2];
    const float* zp = z + ((size_t)b * COUT + o) * Sc;
    size_t yi = ((size_t)b * COUT + o) * Mc + m;
    Y[yi] += w0 * zp[i0] + w1 * zp[i1] + w2 * zp[i2];
}

// ---------------------------------------------------------------------------
// K5: BatchNorm batch stats per channel, folded to y*a + c form.
// stats[o] = a = gamma*rsqrt(var+eps); stats[COUT+o] = c = beta - mu*a
// ---------------------------------------------------------------------------
__global__ void bn_stats_kernel(const float* __restrict__ Y,
                                const float* __restrict__ gamma,
                                const float* __restrict__ beta,
                                float* __restrict__ stats) {
    __shared__ float ssum[256];
    __shared__ float ssq[256];
    const int o = blockIdx.x;
    const int tid = threadIdx.x;
    float s = 0.f, q = 0.f;
    for (int b = 0; b < Bc; ++b) {
        const float* yp = Y + ((size_t)b * COUT + o) * Mc;
        for (int m = tid; m < Mc; m += 256) {
            float v = yp[m];
            s += v;
            q += v * v;
        }
    }
    ssum[tid] = s; ssq[tid] = q;
    __syncthreads();
    for (int st = 128; st > 0; st >>= 1) {
        if (tid < st) { ssum[tid] += ssum[tid + st]; ssq[tid] += ssq[tid + st]; }
        __syncthreads();
    }
    if (tid == 0) {
        const float inv_n = 1.f / (float)(Bc * Mc);
        float mu = ssum[0] * inv_n;
        float var = ssq[0] * inv_n - mu * mu;
        var = var < 0.f ? 0.f : var;
        float a = gamma[o] * rsqrtf(var + 1e-5f);
        stats[o] = a;
        stats[COUT + o] = beta[o] - mu * a;
    }
}

// ---------------------------------------------------------------------------
// K6: in-place affine-BN + LeakyReLU(0.2)
// ---------------------------------------------------------------------------
__global__ void bn_lrelu_kernel(float* __restrict__ Y,
                                const float* __restrict__ stats) {
    size_t t = (size_t)blockIdx.x * blockDim.x + threadIdx.x;
    int o = (int)((t >> 12) & (COUT - 1));     // (t / Mc) % COUT, Mc = 4096
    float v = Y[t] * stats[o] + stats[COUT + o];
    Y[t] = v >= 0.f ? v : 0.2f * v;
}

// ---------------------------------------------------------------------------
extern "C" void kernel_launch(void* const* d_in, const int* in_sizes, int n_in,
                              void* d_out, int out_size, void* d_ws, size_t ws_size,
                              hipStream_t stream) {
    const float* dense_data = (const float*)d_in[0];   // [B][CD][M]
    const int*   dense_idx  = (const int*)  d_in[1];   // [B][M]
    const float* sparse_data= (const float*)d_in[2];   // [B][CS][S]
    const int*   sparse_idx = (const int*)  d_in[3];   // [B][S]
    const float* pcd        = (const float*)d_in[4];   // [B][3][N]
    const float* conv_w     = (const float*)d_in[5];   // [COUT][768]
    const float* bn_gamma   = (const float*)d_in[6];   // [COUT]
    const float* bn_beta    = (const float*)d_in[7];   // [COUT]

    float* yout   = (float*)d_out;                      // [B][COUT][M]
    int*   idxout = (int*)d_out + (size_t)Bc * COUT * Mc;

    // workspace layout (~9.7 MB)
    float* ws    = (float*)d_ws;
    float* dxyz  = ws;                                  // B*3*M
    float* sxyz  = dxyz + (size_t)Bc * 3 * Mc;          // B*3*S
    float* knn_w = sxyz + (size_t)Bc * 3 * Sc;          // B*M*3
    int*   knn_i = (int*)(knn_w + (size_t)Bc * Mc * 3); // B*M*3
    float* zbuf  = (float*)(knn_i + (size_t)Bc * Mc * 3); // B*COUT*S
    float* stats = zbuf + (size_t)Bc * COUT * Sc;       // 2*COUT

    // K0: gather xyz + copy dense_idx out
    {
        int total = Bc * (Mc + Sc);
        gather_xyz_kernel<<<(total + 255) / 256, 256, 0, stream>>>(
            pcd, dense_idx, sparse_idx, dxyz, sxyz, idxout);
    }
    // K1: 3-NN + weights
    knn_kernel<<<dim3(Mc / 256, Bc), 256, 0, stream>>>(dxyz, sxyz, knn_i, knn_w);

    // K2: z[b] = W_sparse(COUT x CS) @ sparse_data[b] (CS x S)
    gemm_wmma_f32<CS, Sc><<<dim3(Sc / 32, COUT / 64, Bc), 128, 0, stream>>>(
        conv_w + CD, sparse_data, zbuf);

    // K3: Y[b] = W_dense(COUT x CD) @ dense_data[b] (CD x M)
    gemm_wmma_f32<CD, Mc><<<dim3(Mc / 32, COUT / 64, Bc), 128, 0, stream>>>(
        conv_w, dense_data, yout);

    // K4: Y += interp(z) via KNN gather (interp-after-conv, linear equivalence)
    interp_add_kernel<<<dim3(Mc / 256, COUT, Bc), 256, 0, stream>>>(
        zbuf, knn_i, knn_w, yout);

    // K5: BN batch stats -> folded scale/shift
    bn_stats_kernel<<<COUT, 256, 0, stream>>>(yout, bn_gamma, bn_beta, stats);

    // K6: in-place normalize + LeakyReLU
    {
        size_t total = (size_t)Bc * COUT * Mc;
        bn_lrelu_kernel<<<(unsigned)(total / 256), 256, 0, stream>>>(yout, stats);
    }
}